// TransOp_expm_62036507623679
// MI455X (gfx1250) — compile-verified
//
#include <hip/hip_runtime.h>
#include <hip/hip_bf16.h>

typedef float v2f __attribute__((ext_vector_type(2)));
typedef float v8f __attribute__((ext_vector_type(8)));

// 3x3 matrix, row i = [ p[i].x, p[i].y, q[i] ]  (v2f pairs encourage v_pk_fma_f32)
struct M3 {
  v2f p[3];
  float q[3];
};

__device__ __forceinline__ v2f splat2(float a) {
  v2f r; r[0] = a; r[1] = a; return r;
}

// C = A * B (row-major 3x3); C must not alias A or B
__device__ __forceinline__ void mm3(const M3& A, const M3& B, M3& C) {
#pragma unroll
  for (int i = 0; i < 3; ++i) {
    const float a0 = A.p[i][0], a1 = A.p[i][1], a2 = A.q[i];
    v2f r = B.p[0] * splat2(a0);
    r = __builtin_elementwise_fma(splat2(a1), B.p[1], r);
    r = __builtin_elementwise_fma(splat2(a2), B.p[2], r);
    C.p[i] = r;
    C.q[i] = fmaf(a0, B.q[0], fmaf(a1, B.q[1], a2 * B.q[2]));
  }
}

// 3x3 matrix exponential: Pade-7 + scaling/squaring, fp32 theta_7 = 3.9257
__device__ __forceinline__ void expm3(const M3& T, M3& E) {
  float n0 = fabsf(T.p[0][0]) + fabsf(T.p[0][1]) + fabsf(T.q[0]);
  float n1 = fabsf(T.p[1][0]) + fabsf(T.p[1][1]) + fabsf(T.q[1]);
  float n2 = fabsf(T.p[2][0]) + fabsf(T.p[2][1]) + fabsf(T.q[2]);
  float nrm = fmaxf(n0, fmaxf(n1, n2));
  int s = 0;
  float sc = 1.0f;
  while (nrm > 3.925724783138660f && s < 32) { nrm *= 0.5f; sc *= 0.5f; ++s; }

  M3 A;
  const v2f scv = splat2(sc);
#pragma unroll
  for (int i = 0; i < 3; ++i) { A.p[i] = T.p[i] * scv; A.q[i] = T.q[i] * sc; }

  M3 A2, A4, A6;
  mm3(A, A, A2);
  mm3(A2, A2, A4);
  mm3(A2, A4, A6);

  const float b0 = 17297280.f, b1 = 8648640.f, b2 = 1995840.f, b3 = 277200.f;
  const float b4 = 25200.f, b5 = 1512.f, b6 = 56.f;  // b7 = 1

  M3 P, V;
#pragma unroll
  for (int i = 0; i < 3; ++i) {
    P.p[i] = __builtin_elementwise_fma(
        splat2(b5), A4.p[i],
        __builtin_elementwise_fma(splat2(b3), A2.p[i], A6.p[i]));
    P.q[i] = fmaf(b5, A4.q[i], fmaf(b3, A2.q[i], A6.q[i]));
    V.p[i] = __builtin_elementwise_fma(
        splat2(b6), A6.p[i],
        __builtin_elementwise_fma(splat2(b4), A4.p[i], A2.p[i] * splat2(b2)));
    V.q[i] = fmaf(b6, A6.q[i], fmaf(b4, A4.q[i], b2 * A2.q[i]));
  }
  P.p[0][0] += b1; P.p[1][1] += b1; P.q[2] += b1;
  V.p[0][0] += b0; V.p[1][1] += b0; V.q[2] += b0;

  M3 U;
  mm3(A, P, U);  // U = A * (A6 + b5 A4 + b3 A2 + b1 I)

  M3 Dm, Nm;
#pragma unroll
  for (int i = 0; i < 3; ++i) {
    Dm.p[i] = V.p[i] - U.p[i]; Nm.p[i] = V.p[i] + U.p[i];
    Dm.q[i] = V.q[i] - U.q[i]; Nm.q[i] = V.q[i] + U.q[i];
  }

  // inverse of Dm via adjugate (scalar cross products)
  const float d0 = Dm.p[0][0], d1 = Dm.p[0][1], d2 = Dm.q[0];
  const float d3 = Dm.p[1][0], d4 = Dm.p[1][1], d5 = Dm.q[1];
  const float d6 = Dm.p[2][0], d7 = Dm.p[2][1], d8 = Dm.q[2];
  const float A00 = d4 * d8 - d5 * d7;
  const float A10 = d5 * d6 - d3 * d8;
  const float A20 = d3 * d7 - d4 * d6;
  const float det = d0 * A00 + d1 * A10 + d2 * A20;
  const float inv = 1.0f / det;
  M3 Iv;
  Iv.p[0][0] = A00 * inv; Iv.p[0][1] = (d2 * d7 - d1 * d8) * inv; Iv.q[0] = (d1 * d5 - d2 * d4) * inv;
  Iv.p[1][0] = A10 * inv; Iv.p[1][1] = (d0 * d8 - d2 * d6) * inv; Iv.q[1] = (d2 * d3 - d0 * d5) * inv;
  Iv.p[2][0] = A20 * inv; Iv.p[2][1] = (d1 * d6 - d0 * d7) * inv; Iv.q[2] = (d0 * d4 - d1 * d3) * inv;

  mm3(Iv, Nm, E);  // E = (V-U)^-1 (V+U)

  for (int k = 0; k < s; ++k) {
    M3 Tq;
    mm3(E, E, Tq);
    E = Tq;
  }
}

// One batch per lane. Per wave: T for 32 batches via 4x v_wmma_f32_16x16x4_f32
// (fp32 GEMM  T^T = psi^T[16x8pad] * c^T[8pad x 16], K split 4+4), then 9
// shuffles redistribute so every lane owns its full 3x3 T; expm + E@x scalar.
__global__ __launch_bounds__(256) void transop_expm_kernel(
    const float* __restrict__ x, const float* __restrict__ c,
    const float* __restrict__ psi, float* __restrict__ out, int B) {
  const int tid  = blockIdx.x * 256 + threadIdx.x;
  const int lane = threadIdx.x & 31;
  const int base = tid - lane;       // first batch of this wave
  const int j    = lane & 15;        // batch-in-half / entry row index
  const bool lo  = (lane < 16);

  float t[9];

  if (base + 32 <= B) {
    // ---- A operand: A[e,m] = psi[m, e] ; e padded 9..15 -> 0 ----
    const bool ev = (j < 9);
    const int m0 = lo ? 0 : 2;  // chunk1 K layout: lanes0-15 hold K=0,1; 16-31 hold K=2,3
    v2f a1, a2;
    a1[0] = ev ? psi[m0 * 9 + j] : 0.0f;
    a1[1] = ev ? psi[(m0 + 1) * 9 + j] : 0.0f;
    a2[0] = (lo && ev) ? psi[4 * 9 + j] : 0.0f;  // chunk2: m = {4,5,pad,pad}
    a2[1] = (lo && ev) ? psi[5 * 9 + j] : 0.0f;

    // ---- B operands: B[m, n] = c[batch, m] for two groups of 16 batches ----
    const float* c0 = c + (size_t)(base + j) * 6;
    const float* c1 = c + (size_t)(base + 16 + j) * 6;
    v2f bl1, bl2, bh1, bh2;
    if (lo) {
      bl1[0] = c0[0]; bl1[1] = c0[1];
      bl2[0] = c0[4]; bl2[1] = c0[5];
      bh1[0] = c1[0]; bh1[1] = c1[1];
      bh2[0] = c1[4]; bh2[1] = c1[5];
    } else {
      bl1[0] = c0[2]; bl1[1] = c0[3];
      bl2[0] = 0.0f;  bl2[1] = 0.0f;
      bh1[0] = c1[2]; bh1[1] = c1[3];
      bh2[0] = 0.0f;  bh2[1] = 0.0f;
    }

    v8f zero = {};
    v8f dlo = __builtin_amdgcn_wmma_f32_16x16x4_f32(false, a1, false, bl1,
                                                    (short)0, zero, false, false);
    dlo = __builtin_amdgcn_wmma_f32_16x16x4_f32(false, a2, false, bl2,
                                                (short)0, dlo, false, false);
    v8f dhi = __builtin_amdgcn_wmma_f32_16x16x4_f32(false, a1, false, bh1,
                                                    (short)0, zero, false, false);
    dhi = __builtin_amdgcn_wmma_f32_16x16x4_f32(false, a2, false, bh2,
                                                (short)0, dhi, false, false);

    // ---- redistribute: lane L<16 -> batch base+L (dlo), L>=16 -> base+L (dhi) ----
#pragma unroll
    for (int r = 0; r < 8; ++r) {
      float vhi = __shfl(dhi[r], j, 32);   // entries 0..7 of batch base+16+j from lane j
      t[r] = lo ? dlo[r] : vhi;
    }
    float e8lo = __shfl(dlo[0], j + 16, 32);  // entry 8 of batch base+j lives in lane j+16
    t[8] = lo ? e8lo : dhi[0];
  } else if (tid < B) {
    // scalar tail (wave-uniform branch keeps EXEC all-ones at WMMA sites)
    const float* crow = c + (size_t)tid * 6;
#pragma unroll
    for (int e = 0; e < 9; ++e) {
      float acc = 0.0f;
#pragma unroll
      for (int m = 0; m < 6; ++m) acc = fmaf(crow[m], psi[m * 9 + e], acc);
      t[e] = acc;
    }
  }

  if (tid < B) {
    M3 T;
#pragma unroll
    for (int i = 0; i < 3; ++i) {
      T.p[i][0] = t[i * 3 + 0];
      T.p[i][1] = t[i * 3 + 1];
      T.q[i]    = t[i * 3 + 2];
    }
    M3 E;
    expm3(T, E);
    const float* xp = x + (size_t)tid * 3;
    const float x0 = xp[0], x1 = xp[1], x2 = xp[2];
    float* op = out + (size_t)tid * 3;
    op[0] = fmaf(E.p[0][0], x0, fmaf(E.p[0][1], x1, E.q[0] * x2));
    op[1] = fmaf(E.p[1][0], x0, fmaf(E.p[1][1], x1, E.q[1] * x2));
    op[2] = fmaf(E.p[2][0], x0, fmaf(E.p[2][1], x1, E.q[2] * x2));
  }
}

extern "C" void kernel_launch(void* const* d_in, const int* in_sizes, int n_in,
                              void* d_out, int out_size, void* d_ws, size_t ws_size,
                              hipStream_t stream) {
  const float* x   = (const float*)d_in[0];   // [B,3,1]
  const float* c   = (const float*)d_in[1];   // [B,6]
  const float* psi = (const float*)d_in[2];   // [6,3,3]
  float* out = (float*)d_out;                 // [B,3,1]
  const int B = in_sizes[1] / 6;
  const int blocks = (B + 255) / 256;
  transop_expm_kernel<<<blocks, 256, 0, stream>>>(x, c, psi, out, B);
}